// SAMsg_64381559767620
// MI455X (gfx1250) — compile-verified
//
#include <hip/hip_runtime.h>
#include <hip/hip_bf16.h>

// ---------------------------------------------------------------------------
// PointNet++ MSG Set Abstraction for gfx1250 (MI455X), wave32.
// Pipeline: FPS -> ball-query+group (3 branches) -> per-branch fused MLP/BN/ReLU
// with layer-wise global-stat passes (recompute, zero activation materialization)
// -> max over nsample -> output.  GEMMs use V_WMMA_F32_16X16X4_F32.
// ---------------------------------------------------------------------------

typedef __attribute__((ext_vector_type(2))) float v2f;
typedef __attribute__((ext_vector_type(8))) float v8f;

#define NBATCH 16
#define NPTS   4096
#define NPOINT 512
#define FEATC  320
#define BN_EPS 1e-5f

__device__ inline v8f wmma_f32(v2f a, v2f b, v8f c) {
  // 8-arg pattern per probe: (neg_a, A, neg_b, B, c_mod, C, reuse_a, reuse_b)
  return __builtin_amdgcn_wmma_f32_16x16x4_f32(false, a, false, b, (short)0, c,
                                               false, false);
}

// ===========================================================================
// 1) Farthest point sampling: one 1024-thread block per batch.
//    Emits new_xyz coords directly into d_out[0 .. 16*512*3).
// ===========================================================================
#define FPS_T 1024
__global__ __launch_bounds__(FPS_T) void fps_kernel(
    const float* __restrict__ x, float* __restrict__ newxyz) {
  int b = blockIdx.x;
  const float* pts = x + (size_t)b * NPTS * 3;
  __shared__ float dist[NPTS];
  __shared__ float rv[FPS_T];
  __shared__ int   ri[FPS_T];
  int t = threadIdx.x;
  for (int i = t; i < NPTS; i += FPS_T) dist[i] = 1e10f;
  __syncthreads();
  int far = 0;
  for (int s = 0; s < NPOINT; ++s) {
    float fx = pts[far * 3 + 0];
    float fy = pts[far * 3 + 1];
    float fz = pts[far * 3 + 2];
    if (t == 0) {
      newxyz[((size_t)b * NPOINT + s) * 3 + 0] = fx;
      newxyz[((size_t)b * NPOINT + s) * 3 + 1] = fy;
      newxyz[((size_t)b * NPOINT + s) * 3 + 2] = fz;
    }
    float best = -1.0f; int bi = 0;
    for (int i = t; i < NPTS; i += FPS_T) {
      float dx = pts[i * 3 + 0] - fx;
      float dy = pts[i * 3 + 1] - fy;
      float dz = pts[i * 3 + 2] - fz;
      float d = dx * dx + dy * dy + dz * dz;
      float dd = fminf(dist[i], d);
      dist[i] = dd;
      if (dd > best) { best = dd; bi = i; }
    }
    rv[t] = best; ri[t] = bi;
    __syncthreads();
    for (int off = FPS_T / 2; off > 0; off >>= 1) {
      if (t < off) {
        float v2 = rv[t + off]; int i2 = ri[t + off];
        if (v2 > rv[t] || (v2 == rv[t] && i2 < ri[t])) { rv[t] = v2; ri[t] = i2; }
      }
      __syncthreads();
    }
    far = ri[0];
    __syncthreads();
  }
}

// ===========================================================================
// 2) Ball query + group: one wave32 per center. Ordered collection of first
//    NS in-radius indices (ascending), padded with the first hit; writes
//    centered coords g[(b*512+j)*NS + s][3].
// ===========================================================================
template <int NS>
__global__ __launch_bounds__(256) void group_kernel(
    const float* __restrict__ x, const float* __restrict__ newxyz,
    float* __restrict__ g, float r2) {
  int w = threadIdx.x >> 5;
  int lane = threadIdx.x & 31;
  int gidx = blockIdx.x * 8 + w;              // grid is exact: 8192/8 blocks
  int b = gidx >> 9;
  const float* pts = x + (size_t)b * NPTS * 3;
  float cx = newxyz[gidx * 3 + 0];
  float cy = newxyz[gidx * 3 + 1];
  float cz = newxyz[gidx * 3 + 2];
  __shared__ int sidx[8][NS];
  int* myidx = sidx[w];
  int count = 0, first = 0;
  bool havefirst = false;
  for (int base = 0; base < NPTS && count < NS; base += 32) {
    int i = base + lane;
    float dx = pts[i * 3 + 0] - cx;
    float dy = pts[i * 3 + 1] - cy;
    float dz = pts[i * 3 + 2] - cz;
    bool hit = (dx * dx + dy * dy + dz * dz) <= r2;
    unsigned mk = (unsigned)__ballot(hit);
    if (!havefirst && mk) { first = base + __ffs(mk) - 1; havefirst = true; }
    if (hit) {
      int pos = count + __popc(mk & ((1u << lane) - 1u));
      if (pos < NS) myidx[pos] = i;
    }
    count += __popc(mk);
  }
  if (count > NS) count = NS;
  for (int s = count + lane; s < NS; s += 32) myidx[s] = first;
  __syncthreads();
  for (int s = lane; s < NS; s += 32) {
    int i = myidx[s];
    size_t o = ((size_t)gidx * NS + s) * 3;
    g[o + 0] = pts[i * 3 + 0] - cx;
    g[o + 1] = pts[i * 3 + 1] - cy;
    g[o + 2] = pts[i * 3 + 2] - cz;
  }
}

// ===========================================================================
// 3) Fused MLP passes. One wave = one (group, 16-row M-tile); the tile flows
//    through all layers in LDS. WMMA f32 16x16x4. Bias dropped (BN-invariant).
// ===========================================================================

// A: LDS, 16 rows x LDA (this wave's tile). W: global [COUT][K] row-major.
template <int K, int LDA>
__device__ inline v8f acc_tile(const float* __restrict__ A,
                               const float* __restrict__ W,
                               int n0, int lane) {
  int kh = (lane >> 4) * 2;
  const float* ar = A + (lane & 15) * LDA;
  const float* wr = W + (size_t)(n0 + (lane & 15)) * K;
  v8f acc = {0, 0, 0, 0, 0, 0, 0, 0};
#pragma unroll
  for (int k = 0; k < K; k += 4) {
    v2f a = {ar[k + kh], ar[k + kh + 1]};
    v2f b = {wr[k + kh], wr[k + kh + 1]};
    acc = wmma_f32(a, b, acc);
  }
  return acc;
}

// Layer 0: K=3 padded to 4 (A pre-padded in LDS, W guarded).
__device__ inline v8f acc_tile_l0(const float* __restrict__ A,
                                  const float* __restrict__ W,
                                  int n0, int lane) {
  int kh = (lane >> 4) * 2;
  int nc = n0 + (lane & 15);
  v2f a = {A[(lane & 15) * 4 + kh], A[(lane & 15) * 4 + kh + 1]};
  v2f b;
  b.x = (kh < 3) ? W[nc * 3 + kh] : 0.0f;
  b.y = (kh + 1 < 3) ? W[nc * 3 + kh + 1] : 0.0f;
  v8f z = {0, 0, 0, 0, 0, 0, 0, 0};
  return wmma_f32(a, b, z);
}

template <int COUT>
__device__ inline void store_act(float* __restrict__ out, v8f acc,
                                 const float* __restrict__ sc,
                                 const float* __restrict__ sh,
                                 int n0, int lane) {
  int col = n0 + (lane & 15);
  int rb = (lane >> 4) * 8;
  float s = sc[col], t = sh[col];
#pragma unroll
  for (int r = 0; r < 8; ++r) {
    float h = acc[r] * s + t;
    out[(rb + r) * COUT + col] = h > 0.0f ? h : 0.0f;
  }
}

__device__ inline void stat_acc(float* __restrict__ ssum,
                                float* __restrict__ ssq,
                                v8f acc, int n0, int lane) {
  int col = n0 + (lane & 15);
  float s1 = 0.0f, s2 = 0.0f;
#pragma unroll
  for (int r = 0; r < 8; ++r) { float h = acc[r]; s1 += h; s2 += h * h; }
  atomicAdd(&ssum[col], s1);
  atomicAdd(&ssq[col], s2);
}

__device__ inline void final_max(int* __restrict__ outcol, v8f acc,
                                 const float* __restrict__ sc,
                                 const float* __restrict__ sh,
                                 int n0, int lane) {
  int col = n0 + (lane & 15);
  float s = sc[col], t = sh[col], m = 0.0f;
#pragma unroll
  for (int r = 0; r < 8; ++r) {
    float h = acc[r] * s + t;
    h = h > 0.0f ? h : 0.0f;
    m = fmaxf(m, h);
  }
  atomicMax(&outcol[col], __float_as_int(m));  // valid: all values >= 0
}

template <int NS, int C0, int C1, int C2, int UPTO, bool FINAL>
__global__ __launch_bounds__(128) void mlp_pass(
    const float* __restrict__ g,
    const float* __restrict__ W0, const float* __restrict__ W1,
    const float* __restrict__ W2,
    const float* __restrict__ sc0, const float* __restrict__ sh0,
    const float* __restrict__ sc1, const float* __restrict__ sh1,
    const float* __restrict__ sc2, const float* __restrict__ sh2,
    float* __restrict__ gsum, float* __restrict__ gsq,
    float* __restrict__ outFeat, int choff) {
  constexpr int MT = NS / 16;
  constexpr int CSTAT = FINAL ? 1 : (UPTO == 1 ? C0 : (UPTO == 2 ? C1 : C2));
  constexpr int CA = (UPTO >= 2) ? C0 : 1;
  constexpr int CB = (UPTO >= 3) ? C1 : 1;
  __shared__ float sg[4][16 * 4];
  __shared__ float bufA[4][16 * CA];
  __shared__ float bufB[4][16 * CB];
  __shared__ float sstat[2 * CSTAT];

  int tid = threadIdx.x, lane = tid & 31, w = tid >> 5;
  if constexpr (!FINAL) {
    for (int c = tid; c < 2 * CSTAT; c += 128) sstat[c] = 0.0f;
    __syncthreads();
  }
  int gw = blockIdx.x * 4 + w;
  int gidx = gw / MT;
  int mt = gw % MT;

  // Load this tile's 16 rows of g, K-padded to 4.
  const float* gp = g + ((size_t)gidx * NS + mt * 16) * 3;
  if (lane < 16) {
    sg[w][lane * 4 + 0] = gp[lane * 3 + 0];
    sg[w][lane * 4 + 1] = gp[lane * 3 + 1];
    sg[w][lane * 4 + 2] = gp[lane * 3 + 2];
    sg[w][lane * 4 + 3] = 0.0f;
  }

  // ---- Layer 0 ----
#pragma unroll
  for (int nt = 0; nt < C0 / 16; ++nt) {
    v8f acc = acc_tile_l0(sg[w], W0, nt * 16, lane);
    if constexpr (UPTO == 1) {
      stat_acc(sstat, sstat + CSTAT, acc, nt * 16, lane);
    } else {
      store_act<C0>(bufA[w], acc, sc0, sh0, nt * 16, lane);
    }
  }
  // ---- Layer 1 ----
  if constexpr (UPTO >= 2) {
#pragma unroll
    for (int nt = 0; nt < C1 / 16; ++nt) {
      v8f acc = acc_tile<C0, C0>(bufA[w], W1, nt * 16, lane);
      if constexpr (UPTO == 2) {
        stat_acc(sstat, sstat + CSTAT, acc, nt * 16, lane);
      } else {
        store_act<C1>(bufB[w], acc, sc1, sh1, nt * 16, lane);
      }
    }
  }
  // ---- Layer 2 ----
  if constexpr (UPTO >= 3) {
#pragma unroll
    for (int nt = 0; nt < C2 / 16; ++nt) {
      v8f acc = acc_tile<C1, C1>(bufB[w], W2, nt * 16, lane);
      if constexpr (FINAL) {
        int* oc = (int*)(outFeat + (size_t)gidx * FEATC + choff);
        final_max(oc, acc, sc2, sh2, nt * 16, lane);
      } else {
        stat_acc(sstat, sstat + CSTAT, acc, nt * 16, lane);
      }
    }
  }
  if constexpr (!FINAL) {
    __syncthreads();
    for (int c = tid; c < CSTAT; c += 128) {
      atomicAdd(&gsum[c], sstat[c]);
      atomicAdd(&gsq[c], sstat[CSTAT + c]);
    }
  }
}

// ===========================================================================
// 4) BN stats -> per-channel affine (scale, shift).  h_bn = h*s + t,
//    s = gamma*rsqrt(var+eps), t = beta - mean*s.
// ===========================================================================
__global__ void finalize_bn(const float* __restrict__ sum,
                            const float* __restrict__ sumsq,
                            const float* __restrict__ gamma,
                            const float* __restrict__ beta,
                            float* __restrict__ scale,
                            float* __restrict__ shift, int C, float invc) {
  int c = threadIdx.x;
  if (c >= C) return;
  float m = sum[c] * invc;
  float v = sumsq[c] * invc - m * m;
  float s = gamma[c] * rsqrtf(v + BN_EPS);
  scale[c] = s;
  shift[c] = beta[c] - m * s;
}

// ===========================================================================
// Host orchestration
// ===========================================================================
struct BranchW {
  const float *W0, *W1, *W2, *gm0, *bt0, *gm1, *bt1, *gm2, *bt2;
};

template <int NS, int C0, int C1, int C2>
static void run_branch(const float* g, const BranchW& p, float* sc0, float* sh0,
                       float* sc1, float* sh1, float* sc2, float* sh2,
                       float* gsum, float* gsq, float* outFeat, int choff,
                       hipStream_t stream) {
  constexpr int MT = NS / 16;
  int blocks = (NBATCH * NPOINT * MT) / 4;
  float invc = 1.0f / ((float)NBATCH * (float)NPOINT * (float)NS);

  hipMemsetAsync(gsum, 0, 256 * sizeof(float), stream);
  mlp_pass<NS, C0, C1, C2, 1, false><<<blocks, 128, 0, stream>>>(
      g, p.W0, p.W1, p.W2, sc0, sh0, sc1, sh1, sc2, sh2, gsum, gsq, outFeat, choff);
  finalize_bn<<<1, 128, 0, stream>>>(gsum, gsq, p.gm0, p.bt0, sc0, sh0, C0, invc);

  hipMemsetAsync(gsum, 0, 256 * sizeof(float), stream);
  mlp_pass<NS, C0, C1, C2, 2, false><<<blocks, 128, 0, stream>>>(
      g, p.W0, p.W1, p.W2, sc0, sh0, sc1, sh1, sc2, sh2, gsum, gsq, outFeat, choff);
  finalize_bn<<<1, 128, 0, stream>>>(gsum, gsq, p.gm1, p.bt1, sc1, sh1, C1, invc);

  hipMemsetAsync(gsum, 0, 256 * sizeof(float), stream);
  mlp_pass<NS, C0, C1, C2, 3, false><<<blocks, 128, 0, stream>>>(
      g, p.W0, p.W1, p.W2, sc0, sh0, sc1, sh1, sc2, sh2, gsum, gsq, outFeat, choff);
  finalize_bn<<<1, 128, 0, stream>>>(gsum, gsq, p.gm2, p.bt2, sc2, sh2, C2, invc);

  mlp_pass<NS, C0, C1, C2, 3, true><<<blocks, 128, 0, stream>>>(
      g, p.W0, p.W1, p.W2, sc0, sh0, sc1, sh1, sc2, sh2, gsum, gsq, outFeat, choff);
}

extern "C" void kernel_launch(void* const* d_in, const int* in_sizes, int n_in,
                              void* d_out, int out_size, void* d_ws,
                              size_t ws_size, hipStream_t stream) {
  const float* x = (const float*)d_in[0];
  auto P = [&](int i, int j, int k) {
    return (const float*)d_in[1 + (i * 3 + j) * 4 + k];
  };
  float* out = (float*)d_out;              // [0,24576): new_xyz ; then features
  float* feat = out + NBATCH * NPOINT * 3;

  // workspace layout (floats)
  float* ws = (float*)d_ws;
  float* g0 = ws;                                          // 16*512*16*3
  float* g1 = g0 + (size_t)NBATCH * NPOINT * 16 * 3;       // 16*512*32*3
  float* g2 = g1 + (size_t)NBATCH * NPOINT * 32 * 3;       // 16*512*128*3
  float* scsh = g2 + (size_t)NBATCH * NPOINT * 128 * 3;    // 9 layers * 256
  auto SC = [&](int i, int j) { return scsh + (i * 3 + j) * 256; };
  auto SH = [&](int i, int j) { return SC(i, j) + 128; };
  float* gsum = scsh + 9 * 256;
  float* gsq = gsum + 128;

  // 1) FPS -> new_xyz (also the query centers)
  fps_kernel<<<NBATCH, FPS_T, 0, stream>>>(x, out);

  // 2) ball query + grouping per branch
  group_kernel<16><<<(NBATCH * NPOINT) / 8, 256, 0, stream>>>(x, out, g0, 0.1f * 0.1f);
  group_kernel<32><<<(NBATCH * NPOINT) / 8, 256, 0, stream>>>(x, out, g1, 0.2f * 0.2f);
  group_kernel<128><<<(NBATCH * NPOINT) / 8, 256, 0, stream>>>(x, out, g2, 0.4f * 0.4f);

  // 3) zero feature output (final pass uses non-negative-float int atomicMax)
  hipMemsetAsync(feat, 0, (size_t)NBATCH * NPOINT * FEATC * sizeof(float), stream);

  // 4) branches
  BranchW b0 = {P(0,0,0), P(0,1,0), P(0,2,0), P(0,0,2), P(0,0,3),
                P(0,1,2), P(0,1,3), P(0,2,2), P(0,2,3)};
  BranchW b1 = {P(1,0,0), P(1,1,0), P(1,2,0), P(1,0,2), P(1,0,3),
                P(1,1,2), P(1,1,3), P(1,2,2), P(1,2,3)};
  BranchW b2 = {P(2,0,0), P(2,1,0), P(2,2,0), P(2,0,2), P(2,0,3),
                P(2,1,2), P(2,1,3), P(2,2,2), P(2,2,3)};

  run_branch<16, 32, 32, 64>(g0, b0, SC(0,0), SH(0,0), SC(0,1), SH(0,1),
                             SC(0,2), SH(0,2), gsum, gsq, feat, 0, stream);
  run_branch<32, 64, 64, 128>(g1, b1, SC(1,0), SH(1,0), SC(1,1), SH(1,1),
                              SC(1,2), SH(1,2), gsum, gsq, feat, 64, stream);
  run_branch<128, 64, 96, 128>(g2, b2, SC(2,0), SH(2,0), SC(2,1), SH(2,1),
                               SC(2,2), SH(2,2), gsum, gsq, feat, 192, stream);
}